// ComplexMaxPool2D_42563125904043
// MI455X (gfx1250) — compile-verified
//
#include <hip/hip_runtime.h>

// ComplexMaxPool2D over (B=8, C2=64, H=320, W=320) fp32, K=2.
// Pure streaming op: 262 MB total traffic, ~11.2 us floor at 23.3 TB/s HBM.
// No data reuse -> direct VGPR path with non-temporal cache policy; no WMMA/TDM role.

typedef float v4f __attribute__((ext_vector_type(4)));
typedef float v2f __attribute__((ext_vector_type(2)));

#define Bsz 8
#define C2  64
#define Cc  32    // complex channels
#define H   320
#define W   320
#define Ho  160
#define Wo  160

// squared-mag -> sqrt, no fma contraction, to mirror reference fp32 semantics
__device__ __forceinline__ float cmag(float re, float im) {
    return __fsqrt_rn(__fadd_rn(__fmul_rn(re, re), __fmul_rn(im, im)));
}

struct CPick { float re, im; };

// argmax over window order (0,0),(0,1),(1,0),(1,1); strict '>' keeps first on ties
__device__ __forceinline__ CPick pick2(float r00, float i00, float r01, float i01,
                                       float r10, float i10, float r11, float i11) {
    float m00 = cmag(r00, i00);
    float m01 = cmag(r01, i01);
    float m10 = cmag(r10, i10);
    float m11 = cmag(r11, i11);
    float bm = m00; CPick p; p.re = r00; p.im = i00;
    if (m01 > bm) { bm = m01; p.re = r01; p.im = i01; }
    if (m10 > bm) { bm = m10; p.re = r10; p.im = i10; }
    if (m11 > bm) { bm = m11; p.re = r11; p.im = i11; }
    return p;
}

__global__ __launch_bounds__(320)
void ComplexMaxPool2D_42563125904043_kernel(const float* __restrict__ x,
                                            float* __restrict__ out) {
    // block = (80, 4): x = wop (pair of output columns), y = sub-row of ho
    const int wop = threadIdx.x;                      // 0..79 -> wo = 2*wop, 2*wop+1
    const int ho  = blockIdx.x * 4 + threadIdx.y;     // 0..159
    const int c   = blockIdx.y;                       // complex channel 0..31
    const int b   = blockIdx.z;                       // batch 0..7

    // input offsets (elements); max ~52.4M fits easily in int32
    const int reBase = (((b * C2 + 2 * c) * H) + 2 * ho) * W + 4 * wop;
    const int imBase = reBase + H * W;                // imag = next channel plane

    // 4x fully-coalesced b128 NT loads (64 B/thread)
    const v4f re0 = __builtin_nontemporal_load((const v4f*)(x + reBase));
    const v4f re1 = __builtin_nontemporal_load((const v4f*)(x + reBase + W));
    const v4f im0 = __builtin_nontemporal_load((const v4f*)(x + imBase));
    const v4f im1 = __builtin_nontemporal_load((const v4f*)(x + imBase + W));

    // output column 2*wop   (input cols 4wop, 4wop+1)
    CPick a = pick2(re0.x, im0.x, re0.y, im0.y, re1.x, im1.x, re1.y, im1.y);
    // output column 2*wop+1 (input cols 4wop+2, 4wop+3)
    CPick bq = pick2(re0.z, im0.z, re0.w, im0.w, re1.z, im1.z, re1.w, im1.w);

    v2f ore, oim;
    ore.x = a.re;  ore.y = bq.re;
    oim.x = a.im;  oim.y = bq.im;

    // 2x coalesced b64 NT stores (16 B/thread)
    const int outRe = ((b * C2 + 2 * c) * Ho + ho) * Wo + 2 * wop;
    __builtin_nontemporal_store(ore, (v2f*)(out + outRe));
    __builtin_nontemporal_store(oim, (v2f*)(out + outRe + Ho * Wo));
}

extern "C" void kernel_launch(void* const* d_in, const int* in_sizes, int n_in,
                              void* d_out, int out_size, void* d_ws, size_t ws_size,
                              hipStream_t stream) {
    (void)in_sizes; (void)n_in; (void)d_ws; (void)ws_size; (void)out_size;
    const float* x = (const float*)d_in[0];
    float* out = (float*)d_out;

    dim3 block(80, 4, 1);             // 320 threads = 10 wave32
    dim3 grid(Ho / 4, Cc, Bsz);       // (40, 32, 8) — no integer divides in kernel
    ComplexMaxPool2D_42563125904043_kernel<<<grid, block, 0, stream>>>(x, out);
}